// K2CTCLoss_74895639707830
// MI455X (gfx1250) — compile-verified
//
#include <hip/hip_runtime.h>
#include <math.h>

// Problem constants (fixed by the reference's setup_inputs()).
#define TT    1600
#define BB    32
#define VV    500
#define LMAXC 100
#define SS    201            // 2*LMAX+1 extended CTC states
#define NEGV  (-1e30f)

#define LOG2E 1.4426950408889634f   // 1/ln(2)
#define LN2   0.6931471805599453f

// ---------------------------------------------------------------------------
// CDNA5 async global->LDS copy (ASYNCcnt-tracked DMA path) with fallbacks.
// ---------------------------------------------------------------------------
#if defined(__has_builtin)
#  if __has_builtin(__builtin_amdgcn_global_load_async_to_lds_b128)
#    define HAVE_ASYNC_LDS_BUILTIN 1
#  endif
#  if __has_builtin(__builtin_amdgcn_s_wait_asynccnt)
#    define HAVE_WAIT_ASYNC_BUILTIN 1
#  endif
#endif

typedef int v4i __attribute__((vector_size(4 * sizeof(int))));
typedef __attribute__((address_space(3))) void lds_void;

__device__ __forceinline__ void async_copy_b128(const void* gsrc, void* lds_dst) {
#if defined(HAVE_ASYNC_LDS_BUILTIN)
  __builtin_amdgcn_global_load_async_to_lds_b128(
      (v4i*)gsrc, (v4i*)lds_dst, 0, 0);
#else
  unsigned lds_off = (unsigned)(unsigned long long)(lds_void*)lds_dst;
  asm volatile("global_load_async_to_lds_b128 %0, %1, off"
               :: "v"(lds_off), "v"(gsrc) : "memory");
#endif
}

__device__ __forceinline__ void wait_async0() {
#if defined(HAVE_WAIT_ASYNC_BUILTIN)
  __builtin_amdgcn_s_wait_asynccnt(0);
#else
  asm volatile("s_wait_asynccnt 0x0" ::: "memory");
#endif
}

// 3-term logsumexp using native v_exp_f32 / v_log_f32 (base-2) units.
// Argument of log2 is in [1,3], so no range issues.
__device__ __forceinline__ float lse3(float a1, float a2, float a3) {
  float m  = fmaxf(a1, fmaxf(a2, a3));
  float s  = __builtin_amdgcn_exp2f((a1 - m) * LOG2E)
           + __builtin_amdgcn_exp2f((a2 - m) * LOG2E)
           + __builtin_amdgcn_exp2f((a3 - m) * LOG2E);
  return m + LN2 * __builtin_amdgcn_logf(s);
}

__device__ __forceinline__ float lse2(float a1, float a2) {
  float m  = fmaxf(a1, a2);
  float s  = __builtin_amdgcn_exp2f((a1 - m) * LOG2E)
           + __builtin_amdgcn_exp2f((a2 - m) * LOG2E);
  return m + LN2 * __builtin_amdgcn_logf(s);
}

// ---------------------------------------------------------------------------
// Alpha recursion: one block per sequence b; thread s owns extended state s.
// Emission rows (V=500 f32 = 2000 B) are double-buffered in LDS via the async
// DMA engine: 125 lanes x b128 covers one row per issue.
// ---------------------------------------------------------------------------
__global__ __launch_bounds__(256)
void ctc_alpha_kernel(const float* __restrict__ log_probs,   // [T,B,V]
                      const int*   __restrict__ targets,     // [B*LMAX]
                      const int*   __restrict__ in_len,      // [B]
                      const int*   __restrict__ tg_len,      // [B]
                      float*       __restrict__ per_seq)     // [B] (ws)
{
  __shared__ float lp[2][512];        // emission row double buffer (500 used)
  __shared__ float alpha[2][SS + 3];  // alpha ping-pong
  __shared__ int   ext[SS];           // extended label per state
  __shared__ int   s_off;             // label-stream offset for this b

  const int b   = blockIdx.x;
  const int tid = threadIdx.x;
  const int L   = tg_len[b];
  const int Tin = in_len[b];

  if (tid == 0) {
    int o = 0;
#pragma unroll 1
    for (int i = 0; i < b; ++i) o += tg_len[i];
    s_off = o;
  }
  __syncthreads();

  if (tid < SS) {
    int e = 0;
    if (tid & 1) {                       // odd states carry labels
      int k   = tid >> 1;
      int idx = s_off + k;
      if (idx > BB * LMAXC - 1) idx = BB * LMAXC - 1;   // clip like reference
      e = (k < L) ? targets[idx] : 0;    // padded tail -> blank id 0
    }
    ext[tid] = e;
  }
  __syncthreads();

  // Per-thread t-invariant registers.
  int   ext_s  = 0;
  bool  skip_s = false;
  float elim   = 0.0f;                 // emission when t >= Tin
  float aself  = NEGV;
  if (tid < SS) {
    ext_s  = ext[tid];
    int em2 = (tid >= 2) ? ext[tid - 2] : -1;
    skip_s = (tid >= 2) && (ext_s != 0) && (ext_s != em2);
    elim   = (ext_s == 0) ? 0.0f : NEGV;
    aself  = (tid == 0) ? 0.0f : NEGV;
    alpha[0][tid] = aself;
  }

  // Pre-issue emission row t=0.
  const size_t rowStride = (size_t)BB * VV;
  if (tid < 125) {
    async_copy_b128(log_probs + (size_t)b * VV + (size_t)tid * 4,
                    &lp[0][tid * 4]);
  }

  int cur = 0;
  for (int t = 0; t < TT; ++t) {
    wait_async0();       // issuing waves: row t has landed in LDS
    __syncthreads();     // publish row t + alpha[cur]; prior reads of the
                         // buffer we are about to overwrite are complete
    if (t + 1 < TT && tid < 125) {
      async_copy_b128(log_probs + (size_t)(t + 1) * rowStride
                                + (size_t)b * VV + (size_t)tid * 4,
                      &lp[(t + 1) & 1][tid * 4]);
    }

    if (tid < SS) {
      float e = (t < Tin) ? lp[t & 1][ext_s] : elim;

      float a1 = aself;
      float a2 = (tid >= 1) ? alpha[cur][tid - 1] : NEGV;
      float a3 = skip_s     ? alpha[cur][tid - 2] : NEGV;

      float r = lse3(a1, a2, a3) + e;

      aself = r;
      alpha[cur ^ 1][tid] = r;
    }
    cur ^= 1;
  }
  __syncthreads();

  if (tid == 0) {
    int sb = 2 * L;
    float al = alpha[cur][sb];
    float ap = alpha[cur][(sb >= 1) ? sb - 1 : 0];
    per_seq[b] = lse2(al, ap);
  }
}

// Deterministic scalar reduction: loss = -sum_b per_seq[b].
__global__ void ctc_reduce_kernel(const float* __restrict__ per_seq,
                                  float* __restrict__ out) {
  if (threadIdx.x == 0 && blockIdx.x == 0) {
    float s = 0.0f;
#pragma unroll 1
    for (int i = 0; i < BB; ++i) s += per_seq[i];
    out[0] = -s;
  }
}

extern "C" void kernel_launch(void* const* d_in, const int* in_sizes, int n_in,
                              void* d_out, int out_size, void* d_ws, size_t ws_size,
                              hipStream_t stream) {
  (void)in_sizes; (void)n_in; (void)out_size; (void)ws_size;
  const float* log_probs = (const float*)d_in[0];
  const int*   targets   = (const int*)  d_in[1];
  const int*   in_len    = (const int*)  d_in[2];
  const int*   tg_len    = (const int*)  d_in[3];
  float*       ws        = (float*)d_ws;         // B per-sequence log-liks

  ctc_alpha_kernel<<<BB, 256, 0, stream>>>(log_probs, targets, in_len, tg_len, ws);
  ctc_reduce_kernel<<<1, 32, 0, stream>>>(ws, (float*)d_out);
}